// GoalPositionModule_50929722196595
// MI455X (gfx1250) — compile-verified
//
#include <hip/hip_runtime.h>
#include <hip/hip_bf16.h>
#include <math.h>

#define NUM_RADIUS 15
#define NUM_ANGLE  48
#define EMB        32
#define NACT       4

typedef __attribute__((ext_vector_type(2))) float v2f;
typedef __attribute__((ext_vector_type(8))) float v8f;

// One wave computes a (<=16)x4 tile of  T(rows x 32) @ Wp(32 x 4) [+ bias]
// using 8x V_WMMA_F32_16X16X4_F32 (fp32-exact), writing rows*4 floats to LDS.
// Branchless operand staging: clamped addresses + value selects (no EXEC
// juggling around the WMMAs).
//
// VGPR layouts per CDNA5 ISA 7.12.2:
//   A 16x4 f32 : lane 0-15 -> M=lane, {v0,v1}={K=0,1}; lane 16-31 -> {K=2,3}
//   B 4x16 f32 : lane 0-15 -> N=lane, {v0,v1}={K=0,1}; lane 16-31 -> {K=2,3}
//   C 16x16 f32: VGPR v, lanes 0-15 -> row v, N=lane; lanes 16-31 -> row v+8
__device__ __forceinline__ void tile_gemm_wmma(const float* __restrict__ T, int rows,
                                               const float* __restrict__ Wp,
                                               const float* __restrict__ bias,
                                               float* __restrict__ outLds) {
  const int lane = threadIdx.x & 31;
  const int half = lane >> 4;   // selects K pair (A/B) and row+8 (C)
  const int mn   = lane & 15;   // M for A, N for B/C
  const bool mv  = (mn < rows);
  const bool nv  = (mn < NACT);
  const int mrow = mv ? mn : (rows - 1);   // clamped A row (in-bounds load)
  const int ncol = mn & (NACT - 1);        // clamped B col (in-bounds load)

  v8f c = {};
#pragma unroll
  for (int k0 = 0; k0 < EMB; k0 += 4) {
    const int ka = k0 + half * 2;
    v2f a, b;
    const float a0 = T[mrow * EMB + ka];
    const float a1 = T[mrow * EMB + ka + 1];
    const float b0 = Wp[ka * NACT + ncol];
    const float b1 = Wp[(ka + 1) * NACT + ncol];
    a.x = mv ? a0 : 0.0f;
    a.y = mv ? a1 : 0.0f;
    b.x = nv ? b0 : 0.0f;
    b.y = nv ? b1 : 0.0f;
    c = __builtin_amdgcn_wmma_f32_16x16x4_f32(false, a, false, b,
                                              (short)0, c, false, false);
  }

  if (nv) {
    const float bb = bias ? bias[mn] : 0.0f;
#pragma unroll
    for (int v = 0; v < 8; ++v) {
      const int row = v + half * 8;
      if (row < rows) outLds[row * NACT + mn] = c[v] + bb;
    }
  }
}

__device__ __forceinline__ float4 row_eval(float dx, float dz, float pose,
                                           const float4* __restrict__ sRW,
                                           const float4* __restrict__ sAW) {
  // radius bucket (accurate math: feeds a discrete index)
  const float radius = sqrtf(dx * dx + dz * dz);
  int ri = (int)(radius / 5.0f);                 // trunc, radius >= 0
  ri = ri > (NUM_RADIUS - 1) ? (NUM_RADIUS - 1) : ri;

  // angle bucket: wrap (90 - atan2*180/pi - pose) into [0, 360)
  const float ang = 90.0f - atan2f(dz, dx) * 57.29577951308232f;
  float d = fmodf(ang - pose, 360.0f);
  if (d < 0.0f) d += 360.0f;
  int ti = (int)(d / 7.5f);
  ti = ti < 0 ? 0 : (ti > (NUM_ANGLE - 1) ? (NUM_ANGLE - 1) : ti);

  const float4 r = sRW[ri];   // LDS b128
  const float4 a = sAW[ti];   // LDS b128
  const float lx = r.x + a.x, ly = r.y + a.y, lz = r.z + a.z, lw = r.w + a.w;

  // log_softmax over 4 — smooth output; fast HW transcendentals are safe here
  const float m  = fmaxf(fmaxf(lx, ly), fmaxf(lz, lw));
  const float s  = __expf(lx - m) + __expf(ly - m) +
                   __expf(lz - m) + __expf(lw - m);
  const float ls = m + __logf(s);
  return make_float4(lx - ls, ly - ls, lz - ls, lw - ls);
}

__global__ void __launch_bounds__(256)
goal_policy_kernel(const float* __restrict__ agent,   // [B,3] x,z,pose
                   const float* __restrict__ goal,    // [B,2]
                   const float* __restrict__ rtab,    // [15,32]
                   const float* __restrict__ atab,    // [48,32]
                   const float* __restrict__ W,       // [64,4]
                   const float* __restrict__ bias,    // [4]
                   float* __restrict__ out,           // [B,4]
                   int Brows) {
  __shared__ __align__(16) float sRW[16 * NACT];        // bias folded in
  __shared__ __align__(16) float sAW[NUM_ANGLE * NACT];

  // --- fused precompute: 4 waves build the 252-float logit LUT via WMMA ---
  const int wave = threadIdx.x >> 5;
  if (wave == 0) {
    tile_gemm_wmma(rtab, NUM_RADIUS, W, bias, sRW);
  } else if (wave <= 3) {
    const int t = wave - 1;                       // 48 = 3 x 16 rows exactly
    tile_gemm_wmma(atab + t * 16 * EMB, 16, W + EMB * NACT, nullptr,
                   sAW + t * 16 * NACT);
  }
  __syncthreads();

  const float4* sRW4 = (const float4*)sRW;
  const float4* sAW4 = (const float4*)sAW;

  // --- streaming phase: 4 rows/thread, all-b128 global traffic ---
  const long tid  = (long)blockIdx.x * blockDim.x + threadIdx.x;
  const long base = tid * 4;
  if (base >= Brows) return;

  if (base + 3 < Brows) {
    const float4* ag = (const float4*)agent + tid * 3;   // 12 floats = 4 rows
    const float4  a0 = ag[0], a1 = ag[1], a2 = ag[2];
    const float4* gp = (const float4*)goal + tid * 2;    // 8 floats = 4 rows
    const float4  g0 = gp[0], g1 = gp[1];

    float4* op = (float4*)out + base;
    op[0] = row_eval(g0.x - a0.x, g0.y - a0.y, a0.z, sRW4, sAW4);
    op[1] = row_eval(g0.z - a0.w, g0.w - a1.x, a1.y, sRW4, sAW4);
    op[2] = row_eval(g1.x - a1.z, g1.y - a1.w, a2.x, sRW4, sAW4);
    op[3] = row_eval(g1.z - a2.y, g1.w - a2.z, a2.w, sRW4, sAW4);
  } else {
    for (long r = base; r < Brows; ++r) {                // tail (B%4 rows)
      const float ax = agent[r * 3 + 0], az = agent[r * 3 + 1],
                  ap = agent[r * 3 + 2];
      const float gx = goal[r * 2 + 0], gz = goal[r * 2 + 1];
      ((float4*)out)[r] = row_eval(gx - ax, gz - az, ap, sRW4, sAW4);
    }
  }
}

extern "C" void kernel_launch(void* const* d_in, const int* in_sizes, int n_in,
                              void* d_out, int out_size, void* d_ws, size_t ws_size,
                              hipStream_t stream) {
  (void)n_in; (void)out_size; (void)d_ws; (void)ws_size;
  const float* agent = (const float*)d_in[0];
  const float* goal  = (const float*)d_in[1];
  const float* rtab  = (const float*)d_in[2];
  const float* atab  = (const float*)d_in[3];
  const float* W     = (const float*)d_in[4];
  const float* bias  = (const float*)d_in[5];
  float*       out   = (float*)d_out;

  const int Brows  = in_sizes[0] / 3;
  const int groups = (Brows + 3) / 4;        // 4 rows per thread
  const int blocks = (groups + 255) / 256;

  goal_policy_kernel<<<blocks, 256, 0, stream>>>(agent, goal, rtab, atab, W,
                                                 bias, out, Brows);
}